// Focal_Loss_13786845020739
// MI455X (gfx1250) — compile-verified
//
#include <hip/hip_runtime.h>

// Focal loss (gamma=1), B rows x 10 classes, fp32. HBM-bound: ~88 MB read.
// CDNA5 data path: per-wave double-buffered GLOBAL_LOAD_ASYNC_TO_LDS_B128
// streaming (perfectly coalesced 512B per instruction, ASYNCcnt-tracked,
// overlapped with exp/log compute), conflict-free LDS row reads, and
// V_WMMA_F32_16X16X4_F32 (B = ones) for the wave-level partial-sum reduce.

typedef __attribute__((ext_vector_type(2))) float v2f;
typedef __attribute__((ext_vector_type(8))) float v8f;

#define TPB 256
#define WAVES_PER_BLOCK (TPB / 32)
#define MAX_BLOCKS 1024
#define NCLS 10
#define ROW_BYTES (NCLS * 4)                 // 40
#define TILE_ROWS 64
#define TILE_BYTES (TILE_ROWS * ROW_BYTES)   // 2560
#define CHUNKS (TILE_BYTES / (32 * 16))      // 5 x b128 per lane per tile

// ---------------- wave / block reduction (WMMA f32 16x16x4) ----------------

__device__ __forceinline__ float wave_reduce(float x) {
#if defined(__has_builtin) && __has_builtin(__builtin_amdgcn_wmma_f32_16x16x4_f32)
    v2f a; a[0] = x;    a[1] = 0.0f;
    v2f b; b[0] = 1.0f; b[1] = 1.0f;
    v8f c = {0.f, 0.f, 0.f, 0.f, 0.f, 0.f, 0.f, 0.f};
    c = __builtin_amdgcn_wmma_f32_16x16x4_f32(false, a, false, b,
                                              (short)0, c, false, false);
    float s = ((c[0] + c[1]) + (c[2] + c[3])) + ((c[4] + c[5]) + (c[6] + c[7]));
    s += __shfl_xor(s, 16, 32);
    return s;
#else
    #pragma unroll
    for (int off = 16; off > 0; off >>= 1) x += __shfl_xor(x, off, 32);
    return x;
#endif
}

__device__ __forceinline__ float block_reduce(float x) {
    __shared__ float smem[WAVES_PER_BLOCK];
    const int lane = threadIdx.x & 31;
    const int wid  = threadIdx.x >> 5;
    x = wave_reduce(x);
    if (lane == 0) smem[wid] = x;
    __syncthreads();
    float r = 0.0f;
    if (wid == 0) {                     // wave-uniform branch: EXEC all-ones
        r = (lane < WAVES_PER_BLOCK) ? smem[lane] : 0.0f;
        r = wave_reduce(r);
    }
    return r;                           // valid in wave 0
}

// ---------------- per-row focal loss (collapsed one-hot math) ----------------

template <typename VP>
__device__ __forceinline__ void load_row(VP r, float x[NCLS]) {
    v2f p0 = r[0], p1 = r[1], p2 = r[2], p3 = r[3], p4 = r[4];
    x[0] = p0[0]; x[1] = p0[1]; x[2] = p1[0]; x[3] = p1[1]; x[4] = p2[0];
    x[5] = p2[1]; x[6] = p3[0]; x[7] = p3[1]; x[8] = p4[0]; x[9] = p4[1];
}

__device__ __forceinline__ float row_focal10(const float x[NCLS], int y) {
    float m = x[0];
    #pragma unroll
    for (int j = 1; j < NCLS; ++j) m = fmaxf(m, x[j]);
    float Z = 0.0f, xy = x[0];
    #pragma unroll
    for (int j = 0; j < NCLS; ++j) {
        Z += __expf(x[j] - m);
        xy = (j == y) ? x[j] : xy;          // constant indices only
    }
    const float ce = (m - xy) + __logf(Z);  // -log p_y (stable)
    const float w  = (xy < m) ? 1.0f : (1.0f - __builtin_amdgcn_rcpf(Z));
    return w * ce;
}

// ---------------- pass 1: per-block partial sums ----------------

__global__ void __launch_bounds__(TPB)
focal_partial(const float* __restrict__ preds,
              const int*   __restrict__ labels,
              float*       __restrict__ blocksums,
              long B) {
    __shared__ __align__(16) char lbuf[WAVES_PER_BLOCK][2][TILE_BYTES];

    const int  lane   = threadIdx.x & 31;
    const int  wv     = threadIdx.x >> 5;
    const long tid    = (long)blockIdx.x * TPB + threadIdx.x;
    const long stride = (long)gridDim.x * TPB;
    const long gw     = (long)blockIdx.x * WAVES_PER_BLOCK + wv;  // global wave
    const long nw     = (long)gridDim.x * WAVES_PER_BLOCK;

    float acc = 0.0f;

    // Async GVS path needs a 32-bit byte offset; fall back to direct loads
    // (ntiles = 0 -> everything goes through the tail loop) if it won't fit.
    const long ntiles =
        ((B * (long)ROW_BYTES) <= 0xFFFF0000L) ? (B / TILE_ROWS) : 0;

    const unsigned laneOff = (unsigned)lane * 16u;

    int  buf = 0;
    long t   = gw;
    if (t < ntiles) {                         // prologue: prefetch first tile
        unsigned l0 = (unsigned)(size_t)&lbuf[wv][0][0] + laneOff;
        unsigned g0 = (unsigned)(t * TILE_BYTES) + laneOff;
        #pragma unroll
        for (int k = 0; k < CHUNKS; ++k) {
            unsigned l = l0 + (unsigned)(k * 512);
            unsigned g = g0 + (unsigned)(k * 512);
            asm volatile("global_load_async_to_lds_b128 %0, %1, %2"
                         :: "v"(l), "v"(g), "s"(preds) : "memory");
        }
    }
    for (; t < ntiles; t += nw) {
        const long tn = t + nw;
        if (tn < ntiles) {                    // prefetch next tile, then wait
            unsigned l0 = (unsigned)(size_t)&lbuf[wv][buf ^ 1][0] + laneOff;
            unsigned g0 = (unsigned)(tn * TILE_BYTES) + laneOff;
            #pragma unroll
            for (int k = 0; k < CHUNKS; ++k) {
                unsigned l = l0 + (unsigned)(k * 512);
                unsigned g = g0 + (unsigned)(k * 512);
                asm volatile("global_load_async_to_lds_b128 %0, %1, %2"
                             :: "v"(l), "v"(g), "s"(preds) : "memory");
            }
            asm volatile("s_wait_asynccnt 0x5" ::: "memory");  // tile t ready
        } else {
            asm volatile("s_wait_asynccnt 0x0" ::: "memory");
        }

        const long rowbase = t * TILE_ROWS;
        #pragma unroll
        for (int rr = 0; rr < 2; ++rr) {
            const int r = lane + rr * 32;     // conflict-free LDS banks
            const v2f* rp = (const v2f*)&lbuf[wv][buf][r * ROW_BYTES];
            float x[NCLS];
            load_row(rp, x);
            acc += row_focal10(x, labels[rowbase + r]);
        }
        // LDS reads must finish before this buffer is re-armed next iteration
        asm volatile("s_wait_dscnt 0x0" ::: "memory");
        buf ^= 1;
    }

    // tail rows (and full fallback when async path disabled)
    for (long i = ntiles * TILE_ROWS + tid; i < B; i += stride) {
        const v2f* rp = (const v2f*)(preds + i * NCLS);
        float x[NCLS];
        load_row(rp, x);
        acc += row_focal10(x, labels[i]);
    }

    const float bs = block_reduce(acc);
    if (threadIdx.x == 0) blocksums[blockIdx.x] = bs;
}

// ---------------- pass 2: deterministic final reduce ----------------

__global__ void __launch_bounds__(TPB)
focal_final(const float* __restrict__ blocksums,
            float*       __restrict__ out,
            int nb, float invB) {
    float acc = 0.0f;
    for (int i = threadIdx.x; i < nb; i += TPB) acc += blocksums[i];
    const float s = block_reduce(acc);
    if (threadIdx.x == 0) out[0] = s * invB;
}

extern "C" void kernel_launch(void* const* d_in, const int* in_sizes, int n_in,
                              void* d_out, int out_size, void* d_ws, size_t ws_size,
                              hipStream_t stream) {
    const float* preds  = (const float*)d_in[0];
    const int*   labels = (const int*)  d_in[1];
    const long   B      = (long)in_sizes[1];   // labels element count

    int nb = MAX_BLOCKS;
    if (ws_size < (size_t)nb * sizeof(float)) {
        nb = (int)(ws_size / sizeof(float));
        if (nb < 1) nb = 1;
    }
    float* blocksums = (float*)d_ws;

    focal_partial<<<nb, TPB, 0, stream>>>(preds, labels, blocksums, B);
    focal_final<<<1, TPB, 0, stream>>>(blocksums, (float*)d_out, nb,
                                       1.0f / (float)B);
}